// DSpec_7808250544781
// MI455X (gfx1250) — compile-verified
//
#include <hip/hip_runtime.h>
#include <math.h>

// ---- static config from the reference ----
#define N_FFT     1024
#define T_FRAMES  622
#define B_BATCH   64
#define L_LEN     160000
#define F_BINS    513                       // N/2+1
#define M_ROWS    (B_BATCH * T_FRAMES)      // 39808 (divisible by 16)
#define NCOLS     1026                      // 2*F (re,im interleaved)
#define NTILE_COL 72                        // padded to 8 waves x 9 tiles (uniform)
#define NT_WAVE   9
#define CPAD      (NTILE_COL * 16)          // 1152 padded basis columns
#define KCHUNK    128
#define SAP       132                       // LDS row stride (floats), conflict pad

#define TAP_BYTES   ((size_t)T_FRAMES * N_FFT * 4)   // 2,547,712
#define IDX_BYTES_A 2560                             // 622*4 rounded up to 256

typedef __attribute__((ext_vector_type(2))) float v2f;
typedef __attribute__((ext_vector_type(8))) float v8f;

#define TWO_PI 6.28318530717958647692f

// ---------------- kernel 1: fractional-shift Hann window per frame ----------------
__global__ __launch_bounds__(256) void win_kernel(const float* __restrict__ wlp,
                                                  const float* __restrict__ stp,
                                                  const float* __restrict__ wpp,
                                                  float* __restrict__ tap,
                                                  int* __restrict__ idxFloor) {
  __shared__ float red[256];
  const int t = blockIdx.x;
  const int tid = threadIdx.x;

  const float wl = fminf(fmaxf(wlp[0], (float)N_FFT / 20.0f), (float)N_FFT);
  const float st = fminf(fmaxf(stp[0], 0.0f), (float)N_FFT);
  const float wp = wpp[0];

  const float frame = st * (float)t;        // frames[t] = t * stride
  const float fl    = floorf(frame);
  const float frac  = frame - fl;
  if (tid == 0) idxFloor[t] = (int)fl;

  const float lo  = floorf(((float)(N_FFT - 1) - wl) * 0.5f);
  const float hi  = ceilf (((float)(N_FFT - 1) + wl) * 0.5f);
  const float off = (wl - (float)N_FFT + 1.0f) * 0.5f;
  const float inv_wl = TWO_PI / wl;

  float vals[4];
  float s = 0.0f;
#pragma unroll
  for (int j = 0; j < 4; ++j) {
    const int n = tid + j * 256;
    const float base = (float)n - frac;
    float v = 0.0f;
    if (!(base >= hi || base <= lo))
      v = 0.5f - 0.5f * cosf((base + off) * inv_wl);
    vals[j] = v;
    s += v;
  }
  red[tid] = s;
  __syncthreads();
  for (int ofs = 128; ofs > 0; ofs >>= 1) {
    if (tid < ofs) red[tid] += red[tid + ofs];
    __syncthreads();
  }
  const float inv_sum = 1.0f / red[0];
#pragma unroll
  for (int j = 0; j < 4; ++j) {
    const int n = tid + j * 256;
    float v = vals[j] * inv_sum;
    if (wp != 1.0f) v = powf(v, wp);
    tap[(size_t)t * N_FFT + n] = v;
  }
}

// -------- kernel 2: DFT basis, K-pair interleaved: bP[p][c] = (row 2p, row 2p+1) --------
// column 2f = cos(2*pi*f*n/N), column 2f+1 = -sin(...)
__global__ __launch_bounds__(256) void basisp_kernel(float* __restrict__ bP) {
  const int p  = blockIdx.x;                // K-pair index: rows n0=2p, n1=2p+1
  const int n0 = 2 * p;
  const int n1 = 2 * p + 1;
  for (int c = threadIdx.x; c < CPAD; c += 256) {
    float v0 = 0.0f, v1 = 0.0f;
    if (c < NCOLS) {
      const int f  = c >> 1;
      const int r0 = (f * n0) & (N_FFT - 1);          // exact range reduction
      const int r1 = (f * n1) & (N_FFT - 1);
      const float a0 = (float)r0 * (TWO_PI / (float)N_FFT);
      const float a1 = (float)r1 * (TWO_PI / (float)N_FFT);
      if (c & 1) { v0 = -sinf(a0); v1 = -sinf(a1); }
      else       { v0 =  cosf(a0); v1 =  cosf(a1); }
    }
    v2f vv; vv.x = v0; vv.y = v1;
    *(v2f*)(bP + 2 * ((size_t)p * CPAD + c)) = vv;    // one b64 store
  }
}

// ---------------- kernel 3: frames x DFT-basis GEMM via f32 WMMA + |.| ----------------
__global__ __launch_bounds__(256) void dft_gemm_kernel(const float* __restrict__ x,
                                                       const float* __restrict__ tap,
                                                       const int* __restrict__ idxFloor,
                                                       const float* __restrict__ basisP,
                                                       float* __restrict__ out) {
  __shared__ float sA[16 * SAP];            // one 16-row x 128-K chunk, shared by 8 waves

  const int m0   = blockIdx.x * 16;
  const int tid  = threadIdx.x;
  const int w    = tid >> 5;                // wave id 0..7 (uniform per wave)
  const int lane = tid & 31;
  const int half = lane >> 4;               // ISA A/B layout: lanes 16-31 hold K+2,K+3
  const int mrow = lane & 15;

  v8f acc[NT_WAVE] = {};                    // 9 persistent f32 accumulators per wave

  for (int c = 0; c < N_FFT / KCHUNK; ++c) {
    __syncthreads();
    // cooperative load: tapered frame chunk -> LDS (16 rows x 128 K)
#pragma unroll
    for (int j = 0; j < 8; ++j) {
      const int e  = tid + j * 256;
      const int r  = e >> 7;
      const int kk = e & 127;
      const int K  = c * KCHUNK + kk;
      const int m  = m0 + r;
      float v = 0.0f;
      if (m < M_ROWS) {
        const int b = m / T_FRAMES;
        const int t = m - b * T_FRAMES;
        const int i = idxFloor[t] + K;
        if ((unsigned)i < (unsigned)L_LEN)
          v = x[(size_t)b * L_LEN + i] * tap[(size_t)t * N_FFT + K];
      }
      sA[r * SAP + kk] = v;
    }
    __syncthreads();

    // consume chunk: K steps of 4 (V_WMMA_F32_16X16X4_F32), fully uniform control flow
    for (int kk = 0; kk < KCHUNK; kk += 4) {
      const v2f a = *(const v2f*)&sA[mrow * SAP + kk + half * 2];   // ds_load_b64
      const int p = c * (KCHUNK / 2) + (kk >> 1) + half;            // K-pair index
      const float* bp = basisP + 2 * ((size_t)p * CPAD + 16 * w + mrow);
#pragma unroll
      for (int ti = 0; ti < NT_WAVE; ++ti) {
        const v2f bv = *(const v2f*)(bp + ti * 256);                // global_load_b64
        acc[ti] = __builtin_amdgcn_wmma_f32_16x16x4_f32(
            false, a, false, bv, (short)0, acc[ti], false, false);
      }
    }
  }

  // magnitude combine (pair adjacent re/im columns via lane shuffle) + write (B,F,T)
#pragma unroll
  for (int ti = 0; ti < NT_WAVE; ++ti) {
    const int gcolbase = (w + ti * 8) * 16;
#pragma unroll
    for (int r = 0; r < 8; ++r) {
      const float v     = acc[ti][r];
      const float other = __shfl_xor(v, 1, 32);   // partner column in (re,im) pair
      const int gcol = gcolbase + mrow;
      const int m    = m0 + half * 8 + r;
      if (((mrow & 1) == 0) && gcol < NCOLS && m < M_ROWS) {
        const float mag = sqrtf(v * v + other * other);
        const int f = gcol >> 1;
        const int b = m / T_FRAMES;
        const int t = m - b * T_FRAMES;
        out[(size_t)b * (F_BINS * T_FRAMES) + (size_t)f * T_FRAMES + t] = mag;
      }
    }
  }
}

extern "C" void kernel_launch(void* const* d_in, const int* in_sizes, int n_in,
                              void* d_out, int out_size, void* d_ws, size_t ws_size,
                              hipStream_t stream) {
  const float* x   = (const float*)d_in[0];
  const float* wlp = (const float*)d_in[1];
  const float* stp = (const float*)d_in[2];
  const float* wpp = (const float*)d_in[3];
  float* out = (float*)d_out;

  char* ws = (char*)d_ws;
  float* tap      = (float*)ws;                                  // T*N f32
  int*   idxFloor = (int*)(ws + TAP_BYTES);                      // T i32
  float* basisP   = (float*)(ws + TAP_BYTES + IDX_BYTES_A);      // 512 * CPAD * 2 f32

  win_kernel<<<T_FRAMES, 256, 0, stream>>>(wlp, stp, wpp, tap, idxFloor);
  basisp_kernel<<<N_FFT / 2, 256, 0, stream>>>(basisP);
  dft_gemm_kernel<<<M_ROWS / 16, 256, 0, stream>>>(x, tap, idxFloor, basisP, out);
}